// UpFIRDn2d_26525718020492
// MI455X (gfx1250) — compile-verified
//
#include <hip/hip_runtime.h>
#include <hip/hip_bf16.h>

// Fused UpFIRDn2d for MI455X (gfx1250).
// One workgroup = one 32x32 output tile of one (b,c) plane.
// Pipeline entirely in LDS:
//   xs (43x43 input patch, async-loaded)  -> A (horizontal 2x-up conv, polyphase)
//   -> Z (vertical 2x-up conv + lrelu*sqrt2) -> W (horizontal stride-2 dn conv, WMMA)
//   -> out (vertical stride-2 dn conv, WMMA, stored to global).
// Down-conv passes run as banded-matrix GEMMs on V_WMMA_F32_16X16X4_F32.

typedef float v8f __attribute__((ext_vector_type(8)));
typedef float v2f __attribute__((ext_vector_type(2)));

#define IMG   128
#define TILE  32
#define SLOPE 0.2f
#define GAIN  1.41421356237f

__global__ __launch_bounds__(256)
void upfirdn2d_fused_kernel(const float* __restrict__ x,
                            const float* __restrict__ bias,
                            const float* __restrict__ uf2d,
                            const float* __restrict__ df2d,
                            float* __restrict__ out)
{
    __shared__ float xs[48 * 48];     // input patch (43x43 valid, zero-padded)
    __shared__ float As[44 * 80];     // after horizontal up-conv (43 x 74 valid)
    __shared__ float Zs[80 * 80];     // after vertical up-conv + lrelu (74x74 valid, rest 0)
    __shared__ float Ws[80 * 32];     // after horizontal dn-conv (74x32 valid, rest 0)
    __shared__ float Ts[11 * 64];     // banded dn-filter slices: T[s][k][m] = fd[4s+k-2m]
    __shared__ float fu[12];          // 1D up filter
    __shared__ float fd[12];          // 1D dn filter

    const int tid  = threadIdx.x;
    const int lane = tid & 31;
    const int wave = tid >> 5;

    const int bc     = blockIdx.x >> 4;       // 0..511  (b*32+c)
    const int tileid = blockIdx.x & 15;
    const int i0 = (tileid >> 2) * TILE;
    const int j0 = (tileid & 3)  * TILE;
    const int r0  = i0 - 5,      c0  = j0 - 5;        // input patch origin
    const int zr0 = 2 * i0 - 5,  zc0 = 2 * j0 - 5;    // upsampled-grid origin
    const float bval = bias[bc & 31];

    // ---- recover 1D filter factors from the 2D outer-product filters ----
    if (tid < 12) {
        fu[tid] = uf2d[tid * 12 + 6] * rsqrtf(uf2d[6 * 12 + 6]);
    } else if (tid < 24) {
        int k = tid - 12;
        fd[k] = df2d[k * 12 + 6] * rsqrtf(df2d[6 * 12 + 6]);
    }

    // ---- zero the padded regions read by the WMMA K-chains ----
    for (int e = tid; e < 80 * 80; e += 256) Zs[e] = 0.0f;
    for (int e = tid; e < 80 * 32; e += 256) Ws[e] = 0.0f;

    // ---- stage 0: input patch -> LDS (async fast path for interior tiles) ----
    const float* xim = x + (long)bc * (IMG * IMG);
    const bool interior = (r0 >= 0) && (r0 + 42 < IMG) && (c0 >= 0) && (c0 + 42 < IMG);

    if (interior) {
        for (int e = tid; e < 43 * 43; e += 256) {
            int rr = e / 43, cc = e % 43;
            const float* gp = xim + (r0 + rr) * IMG + (c0 + cc);
            unsigned long long ga = (unsigned long long)gp;
            // low 32 bits of a generic pointer into LDS == LDS byte address
            unsigned lo = (unsigned)(unsigned long long)(&xs[rr * 48 + cc]);
            asm volatile("global_load_async_to_lds_b32 %0, %1, off"
                         :: "v"(lo), "v"(ga) : "memory");
        }
        asm volatile("s_wait_asynccnt 0x0" ::: "memory");
    } else {
        for (int e = tid; e < 48 * 48; e += 256) {
            int rr = e / 48, cc = e % 48;
            int gr = r0 + rr, gc = c0 + cc;
            float v = 0.0f;
            if (rr < 43 && cc < 43 && gr >= 0 && gr < IMG && gc >= 0 && gc < IMG)
                v = xim[gr * IMG + gc] + bval;   // bias + zero "same" padding
            xs[e] = v;
        }
    }
    __syncthreads();

    if (interior) {   // all 43x43 entries are real pixels: apply bias after async copy
        for (int e = tid; e < 43 * 43; e += 256) {
            int rr = e / 43, cc = e % 43;
            xs[rr * 48 + cc] += bval;
        }
        __syncthreads();
    }

    // ---- build banded dn-filter table (tile-independent) ----
    for (int e = tid; e < 11 * 64; e += 256) {
        int s = e >> 6, k = (e >> 4) & 3, m = e & 15;
        int t = 4 * s + k - 2 * m;
        Ts[e] = (t >= 0 && t < 12) ? fd[t] : 0.0f;
    }

    // ---- stage A: horizontal 2x-up conv (nearest upsample folded via >>1) ----
    for (int e = tid; e < 43 * 74; e += 256) {
        int r = e / 74, zc = e % 74;
        int u = zc0 + zc;
        float acc = 0.0f;
        #pragma unroll
        for (int k = 0; k < 12; ++k) {
            int sx = ((u + k - 5) >> 1) - c0;   // arithmetic shift = floor
            acc = fmaf(xs[r * 48 + sx], fu[k], acc);
        }
        As[r * 80 + zc] = acc;
    }
    __syncthreads();

    // ---- stage B: vertical 2x-up conv + leaky-relu * gain, mask outside 256 grid ----
    for (int e = tid; e < 74 * 74; e += 256) {
        int zr = e / 74, zc = e % 74;
        int u = zr0 + zr;
        float acc = 0.0f;
        #pragma unroll
        for (int k = 0; k < 12; ++k) {
            int sr = ((u + k - 5) >> 1) - r0;
            acc = fmaf(As[sr * 80 + zc], fu[k], acc);
        }
        acc = (acc < 0.0f ? acc * SLOPE : acc) * GAIN;
        int ucol = zc0 + zc;
        bool ok = (u >= 0) && (u < 2 * IMG) && (ucol >= 0) && (ucol < 2 * IMG);
        Zs[zr * 80 + zc] = ok ? acc : 0.0f;     // dn-conv zero padding
    }
    __syncthreads();

    // ---- stage C: horizontal stride-2 dn conv as WMMA banded GEMM ----
    // W[zr][j] = sum_t Z[zr][2j+t] * fd[t]  -> D(16zr x 16j) = Z(16x44) * T(44x16)
    {
        const int m  = lane & 15;
        const int kb = (lane >> 4) * 2;
        for (int t2 = wave; t2 < 10; t2 += 8) {      // 5 zr-blocks x 2 j-blocks
            int zb = t2 >> 1;
            int jb = (t2 & 1) * 16;
            v8f acc = {};
            #pragma unroll
            for (int s = 0; s < 11; ++s) {
                v2f a, b;
                int zrow = zb * 16 + m;
                int zcb  = 2 * jb + 4 * s + kb;
                a.x = Zs[zrow * 80 + zcb];
                a.y = Zs[zrow * 80 + zcb + 1];
                b.x = Ts[s * 64 + kb * 16 + m];        // B(k,n) = T[s][k][n]
                b.y = Ts[s * 64 + (kb + 1) * 16 + m];
                acc = __builtin_amdgcn_wmma_f32_16x16x4_f32(
                          false, a, false, b, (short)0, acc, false, false);
            }
            int hi = (lane >> 4) * 8;
            #pragma unroll
            for (int i = 0; i < 8; ++i)
                Ws[(zb * 16 + i + hi) * 32 + jb + m] = acc[i];
        }
    }
    __syncthreads();

    // ---- stage D: vertical stride-2 dn conv as WMMA banded GEMM, store out ----
    // out[ii][jj] = sum_t W[2ii+t][jj] * fd[t] -> D(16x16) = T^T(16x44) * W(44x16)
    if (wave < 4) {
        const int ib = wave >> 1;
        const int jb = (wave & 1) * 16;
        const int n  = lane & 15;
        const int kb = (lane >> 4) * 2;
        v8f acc = {};
        #pragma unroll
        for (int s = 0; s < 11; ++s) {
            v2f a, b;
            a.x = Ts[s * 64 + kb * 16 + n];            // A(m,k) = T[s][k][m]
            a.y = Ts[s * 64 + (kb + 1) * 16 + n];
            int wr = 32 * ib + 4 * s + kb;
            b.x = Ws[wr * 32 + jb + n];
            b.y = Ws[(wr + 1) * 32 + jb + n];
            acc = __builtin_amdgcn_wmma_f32_16x16x4_f32(
                      false, a, false, b, (short)0, acc, false, false);
        }
        float* op = out + (long)bc * (IMG * IMG);
        int hi = (lane >> 4) * 8;
        #pragma unroll
        for (int i = 0; i < 8; ++i) {
            int ii = ib * 16 + i + hi;
            op[(i0 + ii) * IMG + (j0 + jb + n)] = acc[i];
        }
    }
}

extern "C" void kernel_launch(void* const* d_in, const int* in_sizes, int n_in,
                              void* d_out, int out_size, void* d_ws, size_t ws_size,
                              hipStream_t stream)
{
    (void)in_sizes; (void)n_in; (void)d_ws; (void)ws_size; (void)out_size;
    const float* x    = (const float*)d_in[0];   // (16,32,128,128)
    const float* bias = (const float*)d_in[1];   // (1,32,1,1)
    const float* uf   = (const float*)d_in[2];   // (1,1,12,12)
    const float* df   = (const float*)d_in[3];   // (1,1,12,12)
    float* out        = (float*)d_out;           // (16,32,128,128)

    dim3 grid(512 * 16);   // 512 (b,c) planes x 16 tiles of 32x32
    dim3 block(256);       // 8 wave32
    upfirdn2d_fused_kernel<<<grid, block, 0, stream>>>(x, bias, uf, df, out);
}